// NeighborhoodAttention2D_qk_26207890440784
// MI455X (gfx1250) — compile-verified
//
#include <hip/hip_runtime.h>
#include <hip/hip_bf16.h>

// Problem constants (from reference)
#define NHEADS 8
#define KSZ    7
#define HD     32
#define CDIM   256
#define BB     2
#define HHT    56
#define WWD    56
#define NPIX   (BB * HHT * WWD)   // 6272 pixels
#define BIASW  (2 * KSZ - 1)      // 13

#define LDSPITCH 68               // 64 data floats + 4 pad (272B = 17 x 16B)

typedef float v2f __attribute__((ext_vector_type(2)));
typedef float v8f __attribute__((ext_vector_type(8)));

// Per-lane async copy of 16 bytes global -> LDS, tracked by ASYNCcnt.
__device__ __forceinline__ void async_copy_b128(void* lds_dst, const float* gsrc)
{
    unsigned lds_addr     = (unsigned)(unsigned long long)lds_dst; // low 32b = LDS offset
    unsigned long long ga = (unsigned long long)gsrc;
    asm volatile("global_load_async_to_lds_b128 %0, %1, off"
                 :: "v"(lds_addr), "v"(ga) : "memory");
}

__device__ __forceinline__ void wait_asynccnt_2()
{
    asm volatile("s_wait_asynccnt 0x2" ::: "memory");
}
__device__ __forceinline__ void wait_asynccnt_0()
{
    asm volatile("s_wait_asynccnt 0x0" ::: "memory");
}

// ---------------------------------------------------------------------------
// WMMA fp32 GEMM:  out[N x M] = A[N x K] @ Wm[K x M] + bias[M], then * scale
// Workgroup = 4 waves computes a 64(rows) x 64(cols) tile. Weight panel is
// staged through LDS in 16x64 chunks via async-to-LDS copies, double-buffered
// (prefetch of chunk c+1 overlaps the 16 V_WMMA_F32_16X16X4_F32 of chunk c).
// Fragment layouts per CDNA5 ISA 7.12.2:
//   A (16x4, 32-bit): lanes 0-15 -> rows, v0=K0 v1=K1; lanes 16-31 -> v0=K2 v1=K3
//   B (4x16):         lanes 0-15 -> cols, v0=K0 v1=K1; lanes 16-31 -> v0=K2 v1=K3
//   C/D (16x16):      VGPR g holds M = g + 8*(lane>>4), N = lane&15
// ---------------------------------------------------------------------------
__global__ __launch_bounds__(128)
void gemm_wmma_f32(const float* __restrict__ A,
                   const float* __restrict__ Wm,
                   const float* __restrict__ bias,
                   float* __restrict__ out,
                   int K, int M, float scale)
{
    __shared__ float bsm[2][16][LDSPITCH];   // double-buffered 16(K) x 64(N) panel

    const int tid  = threadIdx.x;
    const int lane = tid & 31;
    const int wave = tid >> 5;      // 0..3
    const int half = lane >> 4;     // 0 or 1
    const int l16  = lane & 15;
    const int row0 = blockIdx.x * 64 + wave * 16;  // this wave's row strip
    const int col0 = blockIdx.y * 64;              // shared col panel

    // staging task: thread copies rows r0 and r0+8 at float-col c0 (16B each)
    const int r0 = tid >> 4;          // 0..7
    const int c0 = (tid & 15) * 4;    // 0..60 step 4

    v8f acc[4];
    #pragma unroll
    for (int t = 0; t < 4; ++t) {
        const float bv = bias[col0 + t * 16 + l16];
        #pragma unroll
        for (int g = 0; g < 8; ++g) acc[t][g] = bv;
    }

    const float* __restrict__ arow = A + (size_t)(row0 + l16) * K;
    const int nchunk = K >> 4;

    // stage chunk 0 into buffer 0
    {
        const float* gp = Wm + (size_t)r0 * M + col0 + c0;
        async_copy_b128(&bsm[0][r0][c0],     gp);
        async_copy_b128(&bsm[0][r0 + 8][c0], gp + (size_t)8 * M);
    }

    for (int c = 0; c < nchunk; ++c) {
        const int cur = c & 1;
        if (c + 1 < nchunk) {
            const int k0n = (c + 1) << 4;
            const float* gp = Wm + (size_t)(k0n + r0) * M + col0 + c0;
            async_copy_b128(&bsm[cur ^ 1][r0][c0],     gp);
            async_copy_b128(&bsm[cur ^ 1][r0 + 8][c0], gp + (size_t)8 * M);
            wait_asynccnt_2();      // chunk c landed; prefetch may stay in flight
        } else {
            wait_asynccnt_0();      // drain
        }
        __syncthreads();

        const int k0 = c << 4;
        if (c + 1 < nchunk)
            __builtin_prefetch(arow + k0 + 16, 0, 1);   // global_prefetch next A chunk

        #pragma unroll
        for (int kk = 0; kk < 16; kk += 4) {
            const int kb = kk + half * 2;
            v2f afrag;
            afrag.x = arow[k0 + kb];
            afrag.y = arow[k0 + kb + 1];
            #pragma unroll
            for (int t = 0; t < 4; ++t) {
                const int n = t * 16 + l16;
                v2f bfrag;
                bfrag.x = bsm[cur][kb][n];
                bfrag.y = bsm[cur][kb + 1][n];
                acc[t] = __builtin_amdgcn_wmma_f32_16x16x4_f32(
                    false, afrag, false, bfrag, (short)0, acc[t], false, false);
            }
        }
        __syncthreads();   // all waves done reading buf[cur] before it is restaged
    }

    #pragma unroll
    for (int t = 0; t < 4; ++t) {
        const int n = col0 + t * 16 + l16;
        #pragma unroll
        for (int g = 0; g < 8; ++g) {
            const int m = row0 + half * 8 + g;
            out[(size_t)m * M + n] = acc[t][g] * scale;
        }
    }
}

// ---------------------------------------------------------------------------
// Neighborhood attention core: one thread per (pixel, head).
// q: [NPIX, 256] (already scaled), kv: [NPIX, 512] (K = cols 0..255, V = 256..511)
// rpb: [8, 13, 13].  out: [NPIX, 256] in (head, hd) order.
// ---------------------------------------------------------------------------
__global__ __launch_bounds__(256)
void natten_core(const float* __restrict__ q,
                 const float* __restrict__ kv,
                 const float* __restrict__ rpb,
                 float* __restrict__ out)
{
    const int tid = blockIdx.x * blockDim.x + threadIdx.x;
    if (tid >= NPIX * NHEADS) return;

    const int head = tid & (NHEADS - 1);
    const int pix  = tid >> 3;
    const int b    = pix / (HHT * WWD);
    const int ij   = pix % (HHT * WWD);
    const int i    = ij / WWD;
    const int j    = ij % WWD;

    int si = i - (KSZ / 2); si = si < 0 ? 0 : si; si = si > (HHT - KSZ) ? (HHT - KSZ) : si;
    int sj = j - (KSZ / 2); sj = sj < 0 ? 0 : sj; sj = sj > (WWD - KSZ) ? (WWD - KSZ) : sj;

    float qreg[HD];
    {
        const float4* qp = (const float4*)(q + (size_t)pix * CDIM + head * HD);
        #pragma unroll
        for (int d4 = 0; d4 < HD / 4; ++d4) {
            float4 v = qp[d4];
            qreg[d4 * 4 + 0] = v.x; qreg[d4 * 4 + 1] = v.y;
            qreg[d4 * 4 + 2] = v.z; qreg[d4 * 4 + 3] = v.w;
        }
    }

    const float* __restrict__ rp = rpb + head * BIASW * BIASW;

    float logit[KSZ * KSZ];
    #pragma unroll
    for (int ki = 0; ki < KSZ; ++ki) {
        const int ih = si + ki;
        const int bi = ih - i + (KSZ - 1);
        #pragma unroll
        for (int kj = 0; kj < KSZ; ++kj) {
            const int iw = sj + kj;
            const int bj = iw - j + (KSZ - 1);
            const float4* kp = (const float4*)(kv +
                ((size_t)((b * HHT + ih) * WWD + iw)) * (2 * CDIM) + head * HD);
            float s = 0.f;
            #pragma unroll
            for (int d4 = 0; d4 < HD / 4; ++d4) {
                float4 v = kp[d4];
                s = fmaf(qreg[d4 * 4 + 0], v.x, s);
                s = fmaf(qreg[d4 * 4 + 1], v.y, s);
                s = fmaf(qreg[d4 * 4 + 2], v.z, s);
                s = fmaf(qreg[d4 * 4 + 3], v.w, s);
            }
            logit[ki * KSZ + kj] = s + rp[bi * BIASW + bj];
        }
    }

    float mx = logit[0];
    #pragma unroll
    for (int t = 1; t < KSZ * KSZ; ++t) mx = fmaxf(mx, logit[t]);
    float ssum = 0.f;
    #pragma unroll
    for (int t = 0; t < KSZ * KSZ; ++t) {
        logit[t] = __expf(logit[t] - mx);
        ssum += logit[t];
    }
    const float inv = 1.0f / ssum;

    float o[HD];
    #pragma unroll
    for (int d = 0; d < HD; ++d) o[d] = 0.f;

    #pragma unroll
    for (int ki = 0; ki < KSZ; ++ki) {
        const int ih = si + ki;
        #pragma unroll
        for (int kj = 0; kj < KSZ; ++kj) {
            const int iw = sj + kj;
            const float4* vp = (const float4*)(kv +
                ((size_t)((b * HHT + ih) * WWD + iw)) * (2 * CDIM) + CDIM + head * HD);
            const float w = logit[ki * KSZ + kj] * inv;
            #pragma unroll
            for (int d4 = 0; d4 < HD / 4; ++d4) {
                float4 v = vp[d4];
                o[d4 * 4 + 0] = fmaf(w, v.x, o[d4 * 4 + 0]);
                o[d4 * 4 + 1] = fmaf(w, v.y, o[d4 * 4 + 1]);
                o[d4 * 4 + 2] = fmaf(w, v.z, o[d4 * 4 + 2]);
                o[d4 * 4 + 3] = fmaf(w, v.w, o[d4 * 4 + 3]);
            }
        }
    }

    float4* op = (float4*)(out + (size_t)pix * CDIM + head * HD);
    #pragma unroll
    for (int d4 = 0; d4 < HD / 4; ++d4) {
        float4 v;
        v.x = o[d4 * 4 + 0]; v.y = o[d4 * 4 + 1];
        v.z = o[d4 * 4 + 2]; v.w = o[d4 * 4 + 3];
        op[d4] = v;
    }
}

// ---------------------------------------------------------------------------
extern "C" void kernel_launch(void* const* d_in, const int* in_sizes, int n_in,
                              void* d_out, int out_size, void* d_ws, size_t ws_size,
                              hipStream_t stream)
{
    (void)in_sizes; (void)n_in; (void)out_size; (void)ws_size;

    const float* x_q   = (const float*)d_in[0];
    const float* x_k   = (const float*)d_in[1];
    const float* Wq    = (const float*)d_in[2];
    const float* bq    = (const float*)d_in[3];
    const float* Wkv   = (const float*)d_in[4];
    const float* bkv   = (const float*)d_in[5];
    const float* rpb   = (const float*)d_in[6];
    const float* Wproj = (const float*)d_in[7];
    const float* bproj = (const float*)d_in[8];
    float* out = (float*)d_out;

    float* ws    = (float*)d_ws;
    float* qbuf  = ws;                                   // NPIX * 256
    float* kvbuf = qbuf  + (size_t)NPIX * CDIM;          // NPIX * 512
    float* abuf  = kvbuf + (size_t)NPIX * 2 * CDIM;      // NPIX * 256

    const float scale = 0.17677669529663687f;            // 32^-0.5

    dim3 blk(128, 1, 1);
    dim3 gq(NPIX / 64, CDIM / 64, 1);        // 98 x 4
    dim3 gkv(NPIX / 64, (2 * CDIM) / 64, 1); // 98 x 8

    // q = (x_q @ Wq + bq) * scale
    gemm_wmma_f32<<<gq, blk, 0, stream>>>(x_q, Wq, bq, qbuf, CDIM, CDIM, scale);
    // kv = x_k @ Wkv + bkv
    gemm_wmma_f32<<<gkv, blk, 0, stream>>>(x_k, Wkv, bkv, kvbuf, CDIM, 2 * CDIM, 1.0f);
    // neighborhood attention core
    natten_core<<<(NPIX * NHEADS + 255) / 256, 256, 0, stream>>>(qbuf, kvbuf, rpb, abuf);
    // out = abuf @ Wproj + bproj
    gemm_wmma_f32<<<gq, blk, 0, stream>>>(abuf, Wproj, bproj, out, CDIM, CDIM, 1.0f);
}